// CEU_GNN_20349555048513
// MI455X (gfx1250) — compile-verified
//
#include <hip/hip_runtime.h>
#include <hip/hip_bf16.h>

typedef __attribute__((ext_vector_type(2))) float v2f;
typedef __attribute__((ext_vector_type(8))) float v8f;

#define N_NODES 100000

// ---------------------------------------------------------------- utilities
__global__ void fill_kernel(float* __restrict__ p, long n, float v) {
    long i = (long)blockIdx.x * blockDim.x + threadIdx.x;
    long stride = (long)gridDim.x * blockDim.x;
    for (; i < n; i += stride) p[i] = v;
}

__global__ void degree_kernel(const int* __restrict__ dst, float* __restrict__ deg, long E) {
    long i = (long)blockIdx.x * blockDim.x + threadIdx.x;
    long stride = (long)gridDim.x * blockDim.x;
    for (; i < E; i += stride) atomicAdd(&deg[dst[i]], 1.0f);
}

__global__ void rsqrt_kernel(float* __restrict__ d, int n) {
    int i = blockIdx.x * blockDim.x + threadIdx.x;
    if (i < n) d[i] = rsqrtf(d[i]);   // deg >= 1 always (self-loop)
}

// ---------------------------------------------------------------- WMMA GEMM
// H[M x NCOLS] = X[M x 128] @ W[128 x NCOLS], fp32 V_WMMA_F32_16X16X4_F32.
// 128 threads (4 waves), 64 rows/block; each wave owns a 16-row strip.
// X tile and (transposed, zero-padded) W strip both staged in LDS so the
// inner K-loop is pure ds_load_b64 + v_wmma at full EXEC: the NCOLS=40
// partial tile is handled entirely at staging/store time.
#define TILE_ROWS 64
#define GEMM_K    128

template <int NCOLS>
__global__ __launch_bounds__(128) void gemm_wmma_f32(
        const float* __restrict__ X, const float* __restrict__ W,
        float* __restrict__ H, int Nrows) {
    constexpr int NT = (NCOLS + 15) / 16;
    __shared__ float sX[TILE_ROWS][GEMM_K + 4];   // +4 pad: 528B rows, 16B aligned
    __shared__ float sWt[16][GEMM_K + 4];         // strip transposed: sWt[col][k]

    const int wave = threadIdx.x >> 5;
    const int lane = threadIdx.x & 31;
    const long rowBase = (long)blockIdx.x * TILE_ROWS;

    // cooperative load of the 64 x 128 X tile (float4 per thread)
    for (int idx = threadIdx.x; idx < TILE_ROWS * (GEMM_K / 4); idx += blockDim.x) {
        int r  = idx >> 5;            // / (GEMM_K/4)
        int c4 = idx & 31;
        long gr = rowBase + r;
        float4 v = {0.f, 0.f, 0.f, 0.f};
        if (gr < Nrows) v = *(const float4*)(X + gr * GEMM_K + c4 * 4);
        *(float4*)&sX[r][c4 * 4] = v;
    }

    const int laneM   = lane & 15;    // A row in strip / B+D col in tile
    const int laneHi  = lane >> 4;    // K pair select {0,1} vs {2,3}
    const int rowStrip = wave * 16;

    for (int nt = 0; nt < NT; ++nt) {
        // stage W strip: sWt[c][k] = W[k][nt*16+c], zero-padded past NCOLS.
        // consecutive threads read consecutive cols -> coalesced; LDS write
        // stride 132 words -> bank step 4 -> conflict-free.
        __syncthreads();
        for (int idx = threadIdx.x; idx < 16 * GEMM_K; idx += blockDim.x) {
            int c = idx & 15;
            int k = idx >> 4;
            int col = nt * 16 + c;
            sWt[c][k] = (col < NCOLS) ? W[k * NCOLS + col] : 0.f;
        }
        __syncthreads();

        v8f acc = {};
        #pragma unroll
        for (int k = 0; k < GEMM_K; k += 4) {
            const int kk = k + 2 * laneHi;            // even -> 8B aligned
            v2f a = *(const v2f*)&sX[rowStrip + laneM][kk];
            v2f b = *(const v2f*)&sWt[laneM][kk];
            acc = __builtin_amdgcn_wmma_f32_16x16x4_f32(
                    /*neg_a=*/false, a, /*neg_b=*/false, b,
                    /*c_mod=*/(short)0, acc, /*reuse_a=*/false, /*reuse_b=*/false);
        }

        // D layout: VGPR j -> M = j + 8*laneHi, N = laneM
        const int col = nt * 16 + laneM;
        if (col < NCOLS) {
            #pragma unroll
            for (int j = 0; j < 8; ++j) {
                long r = rowBase + rowStrip + laneHi * 8 + j;
                if (r < Nrows) H[r * NCOLS + col] = acc[j];
            }
        }
    }
}

// ------------------------------------------------------- edge scatter-add
// One lane = (edge, 4-feature chunk): float4 gather of h[src], scale by
// dinv[s]*dinv[t], 4 float atomicAdds into out[dst]. Working set L2-resident.
template <int F>
__global__ void aggregate_kernel(const int* __restrict__ src, const int* __restrict__ dst,
                                 const float* __restrict__ dinv,
                                 const float* __restrict__ Hf, float* __restrict__ out,
                                 long E) {
    constexpr int chunks = F >> 2;
    const long total = E * chunks;
    long i = (long)blockIdx.x * blockDim.x + threadIdx.x;
    const long stride = (long)gridDim.x * blockDim.x;
    for (; i < total; i += stride) {
        long e = i / chunks;                 // constant divisor -> mul/shift
        int  c = (int)(i - e * chunks);
        int  s = src[e];
        int  t = dst[e];
        float w = dinv[s] * dinv[t];
        float4 hv = *(const float4*)(Hf + (long)s * F + c * 4);
        float* op = out + (long)t * F + c * 4;
        atomicAdd(op + 0, w * hv.x);
        atomicAdd(op + 1, w * hv.y);
        atomicAdd(op + 2, w * hv.z);
        atomicAdd(op + 3, w * hv.w);
    }
}

// self-loop term + bias (+ optional ReLU), fused
template <int F>
__global__ void epilogue_kernel(const float* __restrict__ Hf, const float* __restrict__ dinv,
                                const float* __restrict__ bias, float* __restrict__ out,
                                int N, int do_relu) {
    long total = (long)N * F;
    long i = (long)blockIdx.x * blockDim.x + threadIdx.x;
    const long stride = (long)gridDim.x * blockDim.x;
    for (; i < total; i += stride) {
        long node = i / F;
        int  f    = (int)(i - node * F);
        float di  = dinv[node];
        float v   = out[i] + di * di * Hf[i] + bias[f];
        out[i] = do_relu ? fmaxf(v, 0.f) : v;
    }
}

template <int F>
__global__ void gather_kernel(const float* __restrict__ X, const int* __restrict__ nodes,
                              float* __restrict__ out, int nq) {
    long total = (long)nq * F;
    long i = (long)blockIdx.x * blockDim.x + threadIdx.x;
    const long stride = (long)gridDim.x * blockDim.x;
    for (; i < total; i += stride) {
        long q = i / F;
        int  f = (int)(i - q * F);
        out[i] = X[(long)nodes[q] * F + f];
    }
}

// ---------------------------------------------------------------- driver
extern "C" void kernel_launch(void* const* d_in, const int* in_sizes, int n_in,
                              void* d_out, int out_size, void* d_ws, size_t ws_size,
                              hipStream_t stream) {
    const int*   nodes = (const int*)d_in[0];
    const int*   edge  = (const int*)d_in[1];
    const float* emb   = (const float*)d_in[2];
    const float* W1    = (const float*)d_in[3];
    const float* b1    = (const float*)d_in[4];
    const float* W2    = (const float*)d_in[5];
    const float* b2    = (const float*)d_in[6];
    const float* W3    = (const float*)d_in[7];
    const float* b3    = (const float*)d_in[8];

    const int  nq = in_sizes[0];
    const long E  = (long)in_sizes[1] / 2;
    const int* src = edge;
    const int* dst = edge + E;

    const int N = N_NODES, Hdim = 128, C = 40;

    char* ws = (char*)d_ws;
    float* dinv = (float*)ws;  ws += (size_t)N * sizeof(float);
    float* bufH = (float*)ws;  ws += (size_t)N * Hdim * sizeof(float);
    float* bufA = (float*)ws;  ws += (size_t)N * Hdim * sizeof(float);
    float* bufB = (float*)ws;  ws += (size_t)N * Hdim * sizeof(float);

    const int BS = 256;
    const int GS = 4096;
    dim3 gemmGrid((N + TILE_ROWS - 1) / TILE_ROWS);

    // ---- symmetric normalization: deg (init 1.0 for self-loop) -> dinv
    fill_kernel<<<GS, BS, 0, stream>>>(dinv, N, 1.0f);
    degree_kernel<<<GS, BS, 0, stream>>>(dst, dinv, E);
    rsqrt_kernel<<<(N + BS - 1) / BS, BS, 0, stream>>>(dinv, N);

    // ---- layer 1: relu(agg(emb @ W1) + b1)
    gemm_wmma_f32<128><<<gemmGrid, 128, 0, stream>>>(emb, W1, bufH, N);
    fill_kernel<<<GS, BS, 0, stream>>>(bufA, (long)N * Hdim, 0.f);
    aggregate_kernel<128><<<GS, BS, 0, stream>>>(src, dst, dinv, bufH, bufA, E);
    epilogue_kernel<128><<<GS, BS, 0, stream>>>(bufH, dinv, b1, bufA, N, 1);

    // ---- layer 2
    gemm_wmma_f32<128><<<gemmGrid, 128, 0, stream>>>(bufA, W2, bufH, N);
    fill_kernel<<<GS, BS, 0, stream>>>(bufB, (long)N * Hdim, 0.f);
    aggregate_kernel<128><<<GS, BS, 0, stream>>>(src, dst, dinv, bufH, bufB, E);
    epilogue_kernel<128><<<GS, BS, 0, stream>>>(bufH, dinv, b2, bufB, N, 1);

    // ---- layer 3 (40 output channels, no relu)
    gemm_wmma_f32<40><<<gemmGrid, 128, 0, stream>>>(bufB, W3, bufH, N);
    fill_kernel<<<GS, BS, 0, stream>>>(bufA, (long)N * C, 0.f);
    aggregate_kernel<40><<<GS, BS, 0, stream>>>(src, dst, dinv, bufH, bufA, E);
    epilogue_kernel<40><<<GS, BS, 0, stream>>>(bufH, dinv, b3, bufA, N, 0);

    // ---- final gather of queried nodes
    gather_kernel<40><<<GS, BS, 0, stream>>>(bufA, nodes, (float*)d_out, nq);
}